// BatteriesInteractionBlock_33509334843733
// MI455X (gfx1250) — compile-verified
//
#include <hip/hip_runtime.h>
#include <hip/hip_bf16.h>
#include <math.h>

typedef __attribute__((ext_vector_type(2))) float v2f;
typedef __attribute__((ext_vector_type(8))) float v8f;

#define NPATH 34
#define NGROUP 6
#define MAXK 28          // max k-slots per path group
#define XS_STRIDE 516    // 512 + 4 pad (bank-conflict avoidance, keeps 16B align)
#define Y_STRIDE  17
#define MSG_STRIDE 516

// Path tables (l1 outer, l2 mid, l3 inner — matches python _PATHS order)
__constant__ int c_l1[NPATH] = {0,0,0,0, 1,1,1,1,1,1,1,1,1, 2,2,2,2,2,2,2,2,2,2,2, 3,3,3,3,3,3,3,3,3,3};
__constant__ int c_l2[NPATH] = {0,1,2,3, 0,1,1,1,2,2,2,3,3, 0,1,1,1,2,2,2,2,3,3,3, 0,1,1,2,2,2,3,3,3,3};
__constant__ int c_l3[NPATH] = {0,1,2,3, 1,0,1,2,1,2,3,2,3, 2,1,2,3,0,1,2,3,1,2,3, 3,2,3,1,2,3,0,1,2,3};
// Greedy path groups with sum(2*l3+1) <= MAXK
__constant__ int c_gstart[NGROUP + 1] = {0, 8, 13, 19, 24, 29, 34};
__constant__ int c_ktot[NGROUP]       = {28, 27, 24, 27, 27, 23};

// Device-global CG storage (rebuilt each launch; deterministic)
__device__ float g_cg[NPATH * 343];          // dense padded [p][7][7][7]
__device__ float g_val[NPATH * 7 * 49];      // sparse values per (p,k)
__device__ int   g_code[NPATH * 7 * 49];     // packed (i<<3)|j
__device__ int   g_cnt[NPATH * 7];

static __device__ double factd(int n) {
    double r = 1.0;
    for (int i = 2; i <= n; ++i) r *= (double)i;
    return r;
}

static __device__ double cg_complex(int j1, int m1, int j2, int m2, int j3, int m3) {
    if (m1 + m2 != m3) return 0.0;
    if (j3 < abs(j1 - j2) || j3 > j1 + j2) return 0.0;
    double pre = sqrt((double)(2 * j3 + 1) * factd(j3 + j1 - j2) * factd(j3 - j1 + j2) *
                      factd(j1 + j2 - j3) / factd(j1 + j2 + j3 + 1));
    pre *= sqrt(factd(j3 + m3) * factd(j3 - m3) * factd(j1 - m1) * factd(j1 + m1) *
                factd(j2 - m2) * factd(j2 + m2));
    int kmin = max(0, max(j2 - j3 - m1, j1 + m2 - j3));
    int kmax = min(j1 + j2 - j3, min(j1 - m1, j2 + m2));
    double s = 0.0;
    for (int k = kmin; k <= kmax; ++k) {
        double t = 1.0 / (factd(k) * factd(j1 + j2 - j3 - k) * factd(j1 - m1 - k) *
                          factd(j2 + m2 - k) * factd(j3 - j2 + m1 + k) * factd(j3 - j1 - m2 + k));
        s += (k & 1) ? -t : t;
    }
    return pre * s;
}

// Row 'a' of the real<->complex change-of-basis U(l): at most 2 nonzeros.
static __device__ int urow(int l, int a, int* cols, double* re, double* im) {
    const double s2 = 0.7071067811865475244;
    int m = a - l;
    if (m == 0) { cols[0] = l; re[0] = 1.0; im[0] = 0.0; return 1; }
    if (m > 0) {
        cols[0] = l - m; re[0] = s2;                     im[0] = 0.0;
        cols[1] = l + m; re[1] = (m & 1) ? -s2 : s2;     im[1] = 0.0;
        return 2;
    }
    int mm = -m;
    cols[0] = l - mm; re[0] = 0.0; im[0] = s2;                  // i*s2
    cols[1] = l + mm; re[1] = 0.0; im[1] = (mm & 1) ? s2 : -s2; // -i*(-1)^mm*s2
    return 2;
}

__global__ void cg_dense_kernel() {
    int t = blockIdx.x * blockDim.x + threadIdx.x;
    if (t >= NPATH * 343) return;
    int p = t / 343, rem = t % 343;
    int a = rem / 49, b = (rem / 7) % 7, c = rem % 7;
    int l1 = c_l1[p], l2 = c_l2[p], l3 = c_l3[p];
    float outv = 0.f;
    if (a < 2 * l1 + 1 && b < 2 * l2 + 1 && c < 2 * l3 + 1) {
        int ca[2], cb[2], cc[2];
        double ra[2], ia[2], rb[2], ib[2], rc[2], ic[2];
        int na = urow(l1, a, ca, ra, ia);
        int nb = urow(l2, b, cb, rb, ib);
        int nc = urow(l3, c, cc, rc, ic);
        double accr = 0.0, acci = 0.0;
        for (int x = 0; x < na; ++x)
            for (int y = 0; y < nb; ++y)
                for (int z = 0; z < nc; ++z) {
                    double cg = cg_complex(l1, ca[x] - l1, l2, cb[y] - l2, l3, cc[z] - l3);
                    if (cg == 0.0) continue;
                    double t1r = ra[x] * rb[y] - ia[x] * ib[y];
                    double t1i = ra[x] * ib[y] + ia[x] * rb[y];
                    double t2r = t1r * rc[z] + t1i * ic[z];
                    double t2i = t1i * rc[z] - t1r * ic[z];
                    accr += t2r * cg;
                    acci += t2i * cg;
                }
        outv = (float)(((l1 + l2 + l3) & 1) ? acci : accr);
    }
    g_cg[t] = outv;
}

__global__ void cg_sparse_kernel() {
    int t = blockIdx.x * blockDim.x + threadIdx.x;
    if (t >= NPATH * 7) return;
    int p = t / 7, k = t % 7;
    int l1 = c_l1[p], l2 = c_l2[p], l3 = c_l3[p];
    int cnt = 0;
    if (k < 2 * l3 + 1) {
        for (int i = 0; i < 2 * l1 + 1; ++i)
            for (int j = 0; j < 2 * l2 + 1; ++j) {
                float v = g_cg[p * 343 + i * 49 + j * 7 + k];
                if (fabsf(v) > 1e-7f) {
                    g_val[t * 49 + cnt] = v;
                    g_code[t * 49 + cnt] = (i << 3) | j;
                    ++cnt;
                }
            }
    }
    g_cnt[t] = cnt;
}

__global__ void zero_kernel(float* p, size_t n) {
    size_t i = (size_t)blockIdx.x * blockDim.x + threadIdx.x;
    size_t stride = (size_t)gridDim.x * blockDim.x;
    for (; i < n; i += stride) p[i] = 0.f;
}

// Resolve a group-local k-slot to (path, k). All inputs wave-uniform.
static __device__ __forceinline__ void resolve_slot(int g, int slot, int& pp, int& k) {
    pp = c_gstart[g];
    int base = 0;
    while (true) {
        int d3 = 2 * c_l3[pp] + 1;
        if (slot < base + d3) { k = slot - base; return; }
        base += d3;
        ++pp;
    }
}

// ---------------------------------------------------------------------------
// Edge kernel: 16 edges per block.  Paths processed in 6 groups; per group
// build a[slot][c][e] from sparse CG, then 8 waves pull (slot, d-tile) jobs
// (46..56 per group -> all waves busy) as f32 WMMA 16x16x4 chains (K=32).
// ---------------------------------------------------------------------------
__global__ __launch_bounds__(256) void edge_kernel(
    const float* __restrict__ nodef, const int* __restrict__ eidx,
    const float* __restrict__ evec, const float* __restrict__ tpw,
    float* __restrict__ node_acc, int N, int E)
{
    __shared__ float xs[16 * XS_STRIDE];     // gathered src features [e][c*16+i]
    __shared__ float ysh[16 * Y_STRIDE];     // spherical harmonics  [e][j]
    __shared__ float a_buf[MAXK * 32 * 16];  // a for current group [slot][c][e]
    __shared__ float msg[16 * MSG_STRIDE];   // message accumulator [e][kg*32+d]
    __shared__ int   s_dst[16];

    const int tid = threadIdx.x;
    const int e0  = blockIdx.x * 16;
    const bool full_tile = (e0 + 16 <= E);

    for (int i = tid; i < 16 * MSG_STRIDE; i += 256) msg[i] = 0.f;

    if (!full_tile) {   // zero gather target for OOB edges (tail tile only)
        for (int i = tid; i < 16 * XS_STRIDE; i += 256) xs[i] = 0.f;
        __syncthreads();
    }

    // Gather src node features via async global->LDS B128 copies (ASYNCcnt)
    for (int i = tid; i < 16 * 128; i += 256) {
        int e = i >> 7, off4 = i & 127;
        int ge = e0 + e;
        if (ge < E) {
            int srcn = eidx[ge];
            const float* gptr = nodef + (size_t)srcn * 512 + off4 * 4;
            unsigned lds_off = (unsigned)(uintptr_t)(xs + e * XS_STRIDE + off4 * 4);
            unsigned long long ga = (unsigned long long)(uintptr_t)gptr;
            asm volatile("global_load_async_to_lds_b128 %0, %1, off"
                         :: "v"(lds_off), "v"(ga) : "memory");
        }
    }
    asm volatile("s_wait_asynccnt 0x0" ::: "memory");

    // Spherical harmonics + dst index (one thread per edge)
    if (tid < 16) {
        int ge = e0 + tid;
        float Y[16];
        for (int j = 0; j < 16; ++j) Y[j] = 0.f;
        if (ge < E) {
            float x = evec[ge * 3 + 0], y = evec[ge * 3 + 1], z = evec[ge * 3 + 2];
            float len = fmaxf(sqrtf(x * x + y * y + z * z), 1e-8f);
            x /= len; y /= len; z /= len;
            float x2 = x * x, y2 = y * y, z2 = z * z;
            Y[0]  = 0.28209479177387814f;
            Y[1]  = 0.4886025119029199f * y;
            Y[2]  = 0.4886025119029199f * z;
            Y[3]  = 0.4886025119029199f * x;
            Y[4]  = 1.0925484305920792f * x * y;
            Y[5]  = 1.0925484305920792f * y * z;
            Y[6]  = 0.31539156525252005f * (3.0f * z2 - 1.0f);
            Y[7]  = 1.0925484305920792f * x * z;
            Y[8]  = 0.5462742152960396f * (x2 - y2);
            Y[9]  = 0.5900435899266435f * y * (3.0f * x2 - y2);
            Y[10] = 2.890611442640554f * x * y * z;
            Y[11] = 0.4570457994644658f * y * (5.0f * z2 - 1.0f);
            Y[12] = 0.3731763325901154f * z * (5.0f * z2 - 3.0f);
            Y[13] = 0.4570457994644658f * x * (5.0f * z2 - 1.0f);
            Y[14] = 1.445305721320277f * z * (x2 - y2);
            Y[15] = 0.5900435899266435f * x * (x2 - 3.0f * y2);
            s_dst[tid] = eidx[E + ge];
        } else {
            s_dst[tid] = -1;
        }
        for (int j = 0; j < 16; ++j) ysh[tid * Y_STRIDE + j] = Y[j];
    }
    __syncthreads();

    const int lane = tid & 31;
    const int wave = tid >> 5;
    const int half = lane >> 4;
    const int lid  = lane & 15;

    for (int g = 0; g < NGROUP; ++g) {
        const int ktot = c_ktot[g];

        // Cooperative: a[slot][c][e] = sum_nnz cg * x[e,c,i] * Y[e,j]
        int nel = 512 * ktot;
        for (int idx = tid; idx < nel; idx += 256) {
            int slot = idx >> 9;              // uniform across the wave
            int rem = idx & 511;
            int c = rem >> 4;
            int e = rem & 15;
            int pp, k;
            resolve_slot(g, slot, pp, k);
            int i1off = c_l1[pp] * c_l1[pp];
            int i2off = c_l2[pp] * c_l2[pp];
            int cslot = pp * 7 + k;
            int cnt = g_cnt[cslot];
            const float* vp = g_val + cslot * 49;
            const int*   cp = g_code + cslot * 49;
            const float* xrow = xs + e * XS_STRIDE + c * 16 + i1off;
            const float* yrow = ysh + e * Y_STRIDE + i2off;
            float s = 0.f;
            for (int tt = 0; tt < cnt; ++tt) {
                int code = cp[tt];
                s = fmaf(vp[tt], xrow[code >> 3] * yrow[code & 7], s);
            }
            a_buf[idx] = s;
        }
        __syncthreads();

        // WMMA jobs over the whole group: (slot, d-tile)
        int njobs = ktot * 2;
        for (int job = wave; job < njobs; job += 8) {
            int slot = job >> 1;
            int dt   = job & 1;
            int d    = dt * 16 + lid;
            int pp, k;
            resolve_slot(g, slot, pp, k);
            int l3 = c_l3[pp];
            const float* wbase = tpw + pp * 1024;  // w[c][d]: c=K rows, d=N cols
            v8f acc = {};
#pragma unroll
            for (int c0 = 0; c0 < 32; c0 += 4) {
                int cA = c0 + half * 2;   // lanes16-31 carry K+2 per f32 A/B layout
                v2f A, B;
                A.x = a_buf[(slot * 32 + cA) * 16 + lid];
                A.y = a_buf[(slot * 32 + cA + 1) * 16 + lid];
                B.x = wbase[cA * 32 + d];
                B.y = wbase[(cA + 1) * 32 + d];
                acc = __builtin_amdgcn_wmma_f32_16x16x4_f32(
                    false, A, false, B, (short)0, acc, false, false);
            }
            int kg = l3 * l3 + k;
#pragma unroll
            for (int r = 0; r < 8; ++r) {
                int e = r + half * 8;     // C/D layout: lanes16-31 hold M=8..15
                // paths in the same group may share kg -> LDS atomic (ds_add_f32)
                atomicAdd(&msg[e * MSG_STRIDE + kg * 32 + d], acc[r]);
            }
        }
        __syncthreads();
    }

    // Scatter to per-node accumulator (segment-sum via global f32 atomics)
    for (int i = tid; i < 16 * 512; i += 256) {
        int e   = i >> 9;
        int rem = i & 511;        // = d*16 + kg in output flatten order
        int dd  = rem >> 4;
        int kg  = rem & 15;
        int dn  = s_dst[e];
        if (dn >= 0) {
            atomicAdd(node_acc + (size_t)dn * 512 + rem,
                      msg[e * MSG_STRIDE + kg * 32 + dd]);
        }
    }
}

// ---------------------------------------------------------------------------
// Final linear: out[n,o] = sum_f acc[n,f]*W[o,f] + b[o].
// Block tile: 16 rows x 128 cols, 8 waves, K staged 32 at a time in LDS.
// ---------------------------------------------------------------------------
__global__ __launch_bounds__(256) void linear_kernel(
    const float* __restrict__ acc_in, const float* __restrict__ W,
    const float* __restrict__ bias, float* __restrict__ out, int N)
{
    __shared__ float As[16 * 33];
    __shared__ float Bs[32 * 129];
    const int tid = threadIdx.x;
    const int n0 = blockIdx.x * 16;
    const int o0 = blockIdx.y * 128;
    const int lane = tid & 31, wave = tid >> 5;
    const int half = lane >> 4, lid = lane & 15;

    v8f acc = {};
    for (int f0 = 0; f0 < 512; f0 += 32) {
        for (int i = tid; i < 512; i += 256) {
            int nl = i >> 5, fi = i & 31;
            int n = n0 + nl;
            As[nl * 33 + fi] = (n < N) ? acc_in[(size_t)n * 512 + f0 + fi] : 0.f;
        }
        for (int i = tid; i < 4096; i += 256) {   // Bs[f][o] = W[o][f] (transposed stage)
            int ol = i >> 5, fi = i & 31;
            Bs[fi * 129 + ol] = W[(size_t)(o0 + ol) * 512 + f0 + fi];
        }
        __syncthreads();
        int ob = wave * 16;
#pragma unroll
        for (int c0 = 0; c0 < 32; c0 += 4) {
            int cA = c0 + half * 2;
            v2f A, B;
            A.x = As[lid * 33 + cA];
            A.y = As[lid * 33 + cA + 1];
            B.x = Bs[cA * 129 + ob + lid];
            B.y = Bs[(cA + 1) * 129 + ob + lid];
            acc = __builtin_amdgcn_wmma_f32_16x16x4_f32(
                false, A, false, B, (short)0, acc, false, false);
        }
        __syncthreads();
    }

    int col = o0 + wave * 16 + lid;
    float bv = bias[col];
#pragma unroll
    for (int r = 0; r < 8; ++r) {
        int n = n0 + r + half * 8;
        if (n < N) out[(size_t)n * 512 + col] = acc[r] + bv;
    }
}

extern "C" void kernel_launch(void* const* d_in, const int* in_sizes, int n_in,
                              void* d_out, int out_size, void* d_ws, size_t ws_size,
                              hipStream_t stream)
{
    const float* nodef = (const float*)d_in[0];
    const int*   eidx  = (const int*)d_in[1];
    const float* evec  = (const float*)d_in[2];
    const float* tpw   = (const float*)d_in[3];
    const float* W     = (const float*)d_in[4];
    const float* bias  = (const float*)d_in[5];
    const int N = in_sizes[0] / 512;
    const int E = in_sizes[2] / 3;
    float* node_acc = (float*)d_ws;   // N*512 floats

    cg_dense_kernel<<<(NPATH * 343 + 255) / 256, 256, 0, stream>>>();
    cg_sparse_kernel<<<1, 256, 0, stream>>>();
    zero_kernel<<<2048, 256, 0, stream>>>(node_acc, (size_t)N * 512);

    int ntiles = (E + 15) / 16;
    edge_kernel<<<ntiles, 256, 0, stream>>>(nodef, eidx, evec, tpw, node_acc, N, E);

    dim3 lg((N + 15) / 16, 4);
    linear_kernel<<<lg, 256, 0, stream>>>(node_acc, W, bias, (float*)d_out, N);
}